// GNNSimpleLP_45277545234577
// MI455X (gfx1250) — compile-verified
//
#include <hip/hip_runtime.h>

typedef __attribute__((ext_vector_type(2))) float v2f;
typedef __attribute__((ext_vector_type(8))) float v8f;

#define HID 64   // IN == HID == OUT == 64 in this problem

// ---------------------------------------------------------------------------
// degree / normalization helpers
// ---------------------------------------------------------------------------
__global__ void gnn_fill_zero_kernel(float* __restrict__ p, int n) {
    int i = blockIdx.x * blockDim.x + threadIdx.x;
    if (i < n) p[i] = 0.0f;
}

__global__ void gnn_deg_kernel(const long long* __restrict__ dst,
                               float* __restrict__ deg, int E) {
    int e = blockIdx.x * blockDim.x + threadIdx.x;
    if (e < E) atomicAdd(&deg[dst[e]], 1.0f);
}

__global__ void gnn_isd_kernel(float* __restrict__ deg_isd, int N) {
    int i = blockIdx.x * blockDim.x + threadIdx.x;
    if (i < N) deg_isd[i] = rsqrtf(deg_isd[i] + 1.0f);   // deg incl. self loop
}

// ---------------------------------------------------------------------------
// GEMM: out[N,64] = H[N,64] @ W[64,64]  (+ optional bias)
// Optionally also writes out2 = out * isd[row]^2 (self-loop term, pre-bias).
// One wave computes one 16x16 tile via 16x V_WMMA_F32_16X16X4_F32 over K=64.
// Block = 256 threads = 8 waves -> 32 rows x 64 cols per block. N % 32 == 0.
// ---------------------------------------------------------------------------
__global__ __launch_bounds__(256)
void gnn_gemm64_wmma_kernel(const float* __restrict__ H,
                            const float* __restrict__ W,
                            const float* __restrict__ bias,   // may be null
                            const float* __restrict__ isd,    // used iff out2
                            float* __restrict__ out,
                            float* __restrict__ out2,         // may be null
                            int N) {
    const int lane = threadIdx.x & 31;
    const int wave = threadIdx.x >> 5;                 // 0..7
    const int rowBase = blockIdx.x * 32 + (wave >> 2) * 16;
    const int colBase = (wave & 3) * 16;
    if (rowBase >= N) return;                          // wave-uniform

    const int r  = lane & 15;   // A: row-in-tile; B/C/D: col-in-tile
    const int kh = lane >> 4;   // K-half selector (ISA f32 16x4 layout)

    const float* aptr = H + (size_t)(rowBase + r) * HID + 2 * kh;
    const float* bptr = W + (size_t)(2 * kh) * HID + colBase + r;

    v8f acc = {};
    #pragma unroll
    for (int k = 0; k < HID; k += 4) {
        v2f a, b;
        a.x = aptr[k];                      // H[row][k + 2*kh]     (float2 load)
        a.y = aptr[k + 1];                  // H[row][k + 2*kh + 1]
        b.x = bptr[(size_t)k * HID];        // W[k + 2*kh    ][col]
        b.y = bptr[(size_t)k * HID + HID];  // W[k + 2*kh + 1][col]
        acc = __builtin_amdgcn_wmma_f32_16x16x4_f32(
                  false, a, false, b, (short)0, acc, false, false);
    }

    const int col = colBase + r;
    const float bv = bias ? bias[col] : 0.0f;
    #pragma unroll
    for (int m = 0; m < 8; ++m) {
        const int grow = rowBase + m + 8 * kh;         // C/D VGPR layout
        const float v = acc[m] + bv;
        out[(size_t)grow * HID + col] = v;
        if (out2) {
            const float s = isd[grow];
            out2[(size_t)grow * HID + col] = v * s * s; // bias==null on this path
        }
    }
}

// ---------------------------------------------------------------------------
// agg[i] = m[i] * isd[row]^2   (self-loop init when it can't be fused)
// ---------------------------------------------------------------------------
__global__ void gnn_selfloop_init_kernel(const float* __restrict__ m,
                                         const float* __restrict__ isd,
                                         float* __restrict__ agg, size_t n) {
    size_t i = (size_t)blockIdx.x * blockDim.x + threadIdx.x;
    if (i < n) {
        const float s = isd[i >> 6];
        agg[i] = m[i] * s * s;
    }
}

// ---------------------------------------------------------------------------
// Edge scatter: agg[dst] += m[src] * isd[src]*isd[dst]
// 16 threads per edge, float4 per thread, 4 fp32 atomics. L2-resident.
// ---------------------------------------------------------------------------
__global__ __launch_bounds__(256)
void gnn_scatter_kernel(const float* __restrict__ m,
                        const long long* __restrict__ src,
                        const long long* __restrict__ dst,
                        const float* __restrict__ isd,
                        float* __restrict__ agg, int E) {
    const int t = blockIdx.x * blockDim.x + threadIdx.x;
    const int e = t >> 4;
    if (e >= E) return;
    const int f = (t & 15) * 4;

    if ((t & 15) == 0) {                 // keep the index stream ahead
        __builtin_prefetch(&src[e + 4096], 0, 0);
        __builtin_prefetch(&dst[e + 4096], 0, 0);
    }

    const long long s = src[e];
    const long long d = dst[e];
    const float norm = isd[s] * isd[d];

    const float4 mv = *(const float4*)(m + (size_t)s * HID + f);
    float* o = agg + (size_t)d * HID + f;
    atomicAdd(o + 0, mv.x * norm);
    atomicAdd(o + 1, mv.y * norm);
    atomicAdd(o + 2, mv.z * norm);
    atomicAdd(o + 3, mv.w * norm);
}

// ---------------------------------------------------------------------------
// h = relu(agg + b)  (in place)
// ---------------------------------------------------------------------------
__global__ void gnn_relu_bias_kernel(float* __restrict__ h,
                                     const float* __restrict__ bias, size_t n) {
    size_t i = (size_t)blockIdx.x * blockDim.x + threadIdx.x;
    if (i < n) {
        const float v = h[i] + bias[i & (HID - 1)];
        h[i] = fmaxf(v, 0.0f);
    }
}

// ---------------------------------------------------------------------------
extern "C" void kernel_launch(void* const* d_in, const int* in_sizes, int n_in,
                              void* d_out, int out_size, void* d_ws, size_t ws_size,
                              hipStream_t stream) {
    const float*     x   = (const float*)d_in[0];
    const long long* ei  = (const long long*)d_in[1];   // int64 [2,E] flat
    const float*     W1  = (const float*)d_in[2];
    const float*     b1  = (const float*)d_in[3];
    const float*     W2  = (const float*)d_in[4];
    const float*     b2  = (const float*)d_in[5];
    const float*     Wp  = (const float*)d_in[6];
    const float*     bp  = (const float*)d_in[7];
    float*           out = (float*)d_out;

    const int N = in_sizes[0] / HID;
    const int E = in_sizes[1] / 2;
    const long long* src = ei;
    const long long* dstv = ei + E;

    // workspace: isd[N] | bufA[N*64] | bufB[N*64]   (~52 MB)
    float* isd  = (float*)d_ws;
    float* bufA = isd + N;
    float* bufB = bufA + (size_t)N * HID;

    const int TPB = 256;
    const size_t elems = (size_t)N * HID;
    const int nBlk    = (N + TPB - 1) / TPB;
    const int eBlk    = (E + TPB - 1) / TPB;
    const int eltBlk  = (int)((elems + TPB - 1) / TPB);
    const int scatBlk = (int)(((size_t)E * 16 + TPB - 1) / TPB);
    const int gemmBlk = (N + 31) / 32;

    // normalization: deg -> isd
    gnn_fill_zero_kernel<<<nBlk, TPB, 0, stream>>>(isd, N);
    gnn_deg_kernel<<<eBlk, TPB, 0, stream>>>(dstv, isd, E);
    gnn_isd_kernel<<<nBlk, TPB, 0, stream>>>(isd, N);

    // layer 1: m=x@W1 -> bufA, fused agg-init -> bufB; scatter; relu(+b1)
    gnn_gemm64_wmma_kernel<<<gemmBlk, TPB, 0, stream>>>(x, W1, nullptr, isd,
                                                        bufA, bufB, N);
    gnn_scatter_kernel<<<scatBlk, TPB, 0, stream>>>(bufA, src, dstv, isd, bufB, E);
    gnn_relu_bias_kernel<<<eltBlk, TPB, 0, stream>>>(bufB, b1, elems);

    // layer 2: m=h1@W2 -> bufA (bufB still live during GEMM -> separate init)
    gnn_gemm64_wmma_kernel<<<gemmBlk, TPB, 0, stream>>>(bufB, W2, nullptr, nullptr,
                                                        bufA, nullptr, N);
    gnn_selfloop_init_kernel<<<eltBlk, TPB, 0, stream>>>(bufA, isd, bufB, elems);
    gnn_scatter_kernel<<<scatBlk, TPB, 0, stream>>>(bufA, src, dstv, isd, bufB, E);
    gnn_relu_bias_kernel<<<eltBlk, TPB, 0, stream>>>(bufB, b2, elems);

    // head: out = h2@Wp + bp   (bias fused in GEMM epilogue)
    gnn_gemm64_wmma_kernel<<<gemmBlk, TPB, 0, stream>>>(bufB, Wp, bp, nullptr,
                                                        out, nullptr, N);
}